// PSAMask_3882650436038
// MI455X (gfx1250) — compile-verified
//
#include <hip/hip_runtime.h>
#include <stdint.h>

// PSAMask "collect": out[n][i*65+j][h][w] = x[n][(i-h+64)*129 + (j-w+64)][h][w]
// Pure permutation (no FLOPs); bandwidth-bound (~572 MB total -> ~24.5us at 23.3 TB/s).
// Per-(n,h,i) tile:
//   1) predicated GLOBAL_LOAD_ASYNC_TO_LDS_B32 of the used diagonal band,
//      shear-placed directly into a dense 65x65 LDS tile (per-lane LDS dest),
//   2) s_wait_asynccnt 0 + barrier,
//   3) one TDM tensor_store_from_lds (2D descriptor, row stride 4225 floats).

#define NB   4
#define HH   65
#define WW   65
#define MH   129
#define MWID 129
#define HALF 64
#define HW   (HH * WW)    // 4225
#define MHW  (MH * MWID)  // 16641

typedef uint32_t u32x4 __attribute__((ext_vector_type(4)));
typedef uint32_t u32x8 __attribute__((ext_vector_type(8)));

__global__ __launch_bounds__(256) void psa_collect_tdm(const float* __restrict__ x,
                                                       float* __restrict__ out) {
    __shared__ float tile[HW];  // dense 65x65 output tile, row-major [j][w]

    const int b = blockIdx.x;          // b = ((n*65) + h)*65 + i
    const int i = b % HH;
    const int h = (b / HH) % HH;
    const int n = b / (HH * HH);

    const int  mh     = i - h + HALF;  // [0,128], always in range
    const long inbase = ((long)n * MHW + (long)mh * MWID) * HW + (long)h * WW;
    // element index of x[n][mh*129 + mw][h][w] == inbase + mw*HW + w

    const float*   xblk     = x + inbase;                       // uniform -> SGPR pair
    const uint32_t lds_base = (uint32_t)(uintptr_t)(&tile[0]);  // flat low 32 = LDS offset

    // Async global->LDS copy of the used band (j = mw + w - 64 in [0,65)).
    // Global side coalesced along w; LDS dest sheared per lane (dword delta 66
    // across lanes -> bank delta 2 -> conflict-free).
    for (int e = threadIdx.x; e < MWID * WW; e += 256) {
        const int mw = e / WW;
        const int w  = e - mw * WW;
        const int j  = mw + w - HALF;
        if ((unsigned)j < (unsigned)HH) {
            const uint32_t goff  = (uint32_t)(mw * HW + w) * 4u;            // <= ~2.2 MB
            const uint32_t laddr = lds_base + (uint32_t)(j * WW + w) * 4u;
            asm volatile("global_load_async_to_lds_b32 %0, %1, %2"
                         :
                         : "v"(laddr), "v"(goff), "s"(xblk)
                         : "memory");
        }
    }
    // Ensure this wave's async LDS writes have landed before signaling barrier.
    asm volatile("s_wait_asynccnt 0x0" ::: "memory");
    __syncthreads();

    // One TDM store per workgroup: dense 65x65 tile from LDS to
    // out[n][i*65 + j][h][w], row stride 4225 floats.
    if (threadIdx.x == 0) {
        const uint64_t gaddr =
            (uint64_t)(uintptr_t)out +
            4ull * (uint64_t)(((long)n * HW + (long)i * WW) * (long)HW + (long)h * WW);

        // D# group 0 (128b): count=1 | lds_addr | global_addr[56:0] | type=2
        u32x4 g0;
        g0.x = 0x1u;
        g0.y = lds_base;
        g0.z = (uint32_t)gaddr;
        g0.w = (uint32_t)(gaddr >> 32) | 0x80000000u;

        // D# group 1 (256b)
        u32x8 g1;
        g1[0] = 0x2u << 16;            // workgroup_mask=0, data_size=2 (4 bytes)
        g1[1] = (uint32_t)WW << 16;    // tensor_dim0[15:0] = 65 (atomic_barrier_addr=0)
        g1[2] = (uint32_t)HH << 16;    // tensor_dim1[15:0] = 65 (tensor_dim0 hi = 0)
        g1[3] = (uint32_t)WW << 16;    // tile_dim0 = 65      (tensor_dim1 hi = 0)
        g1[4] = (uint32_t)HH;          // tile_dim1 = 65, tile_dim2 = 0 (2D)
        g1[5] = (uint32_t)HW;          // tensor_dim0_stride = 4225 (data_size units)
        g1[6] = 0u;                    // dim0_stride hi, dim1_stride lo
        g1[7] = 0u;                    // dim1_stride hi

        asm volatile("tensor_store_from_lds %0, %1"
                     :
                     : "s"(g0), "s"(g1)
                     : "memory");
        __builtin_amdgcn_s_wait_tensorcnt(0);
    }
}

extern "C" void kernel_launch(void* const* d_in, const int* in_sizes, int n_in,
                              void* d_out, int out_size, void* d_ws, size_t ws_size,
                              hipStream_t stream) {
    (void)in_sizes; (void)n_in; (void)out_size; (void)d_ws; (void)ws_size;
    const float* x   = (const float*)d_in[0];
    float*       out = (float*)d_out;

    const int nblocks = NB * HH * HH;  // 16900 tiles: one per (n, h, i)
    psa_collect_tdm<<<nblocks, 256, 0, stream>>>(x, out);
}